// Yolov1Loss_22780506538099
// MI455X (gfx1250) — compile-verified
//
#include <hip/hip_runtime.h>

// YOLOv1 loss for MI455X (gfx1250, wave32).
// Roofline: 2 x 802816 cells x 120 B = 193 MB single pass, ~0.1 GFLOP total
// -> pure HBM-bound (~8 us @ 23.3 TB/s; working set ~fits 192 MB L2 on replay).
// Stage 1: 784 blocks x 128 threads; each block streams 1024 contiguous cells as
//          8 chunks of 128 cells, double-buffered in LDS via async global->LDS
//          b128 copies (ASYNCcnt), compute reads LDS conflict-free.
// Stage 2: single-block deterministic reduction of the 784 partials.
// Cross-wave block reduction uses one V_WMMA_F32_16X16X4_F32 (B = ones).

#define NCELLS            (16384 * 49)       // 802816
#define S1_BLOCK          128                // 4 waves
#define CHUNK_CELLS       128
#define CHUNKS_PER_BLOCK  8
#define S1_GRID           (NCELLS / (CHUNK_CELLS * CHUNKS_PER_BLOCK))  // 784 exact
#define S2_BLOCK          256

#define CHUNK_F4          1920               // float4 slots per chunk (pred+targ)
#define CHUNK_F4_PRED     960                // first 960 = pred, next 960 = targ
#define ASYNC_ITERS       (CHUNK_F4 / S1_BLOCK)  // 15 async b128 issues per wave

typedef __attribute__((ext_vector_type(2))) float v2f;
typedef __attribute__((ext_vector_type(8))) float v8f;

#if __has_builtin(__builtin_amdgcn_global_load_async_to_lds_b128) && \
    __has_builtin(__builtin_amdgcn_s_wait_asynccnt)
#define USE_ASYNC_LDS 1
// The builtin wants pointers to 16-byte int vectors with AS1/AS3 pointees
// (clang prints AS1 as "__device__"). Build them from raw addresses.
typedef int v4i __attribute__((vector_size(16)));
typedef __attribute__((address_space(1))) v4i as1_v4i;  // global pointee
typedef __attribute__((address_space(3))) v4i as3_v4i;  // LDS pointee (32-bit ptr)
#else
#define USE_ASYNC_LDS 0
#endif

// ---------------- block reduction: shuffles + one f32 WMMA ----------------
template <int NWAVES>
__device__ __forceinline__ float block_reduce(float acc) {
    #pragma unroll
    for (int off = 16; off > 0; off >>= 1)
        acc += __shfl_xor(acc, off, 32);

    __shared__ float wsum[8];
    const int lane = threadIdx.x & 31;
    const int wid  = threadIdx.x >> 5;
    if (lane == 0) wsum[wid] = acc;
    if (NWAVES < 8 && threadIdx.x >= NWAVES && threadIdx.x < 8)
        wsum[threadIdx.x] = 0.0f;
    __syncthreads();

    float blocksum = 0.0f;
    if (wid == 0) {  // wave-uniform: EXEC all-ones for the WMMA
#if __has_builtin(__builtin_amdgcn_wmma_f32_16x16x4_f32)
        // A (16x4 f32, 2 VGPRs): A[m][0]=lane m v0, A[m][1]=lane m v1,
        // A[m][2]=lane m+16 v0, A[m][3]=lane m+16 v1. Rows 0-1 hold the 8 sums,
        // B = ones -> D[0][n]=s0..s3, D[1][n]=s4..s7; total = c[0]+c[1] (lanes 0-15).
        v2f a; a[0] = 0.0f; a[1] = 0.0f;
        if (lane == 0)       { a[0] = wsum[0]; a[1] = wsum[1]; }
        else if (lane == 16) { a[0] = wsum[2]; a[1] = wsum[3]; }
        else if (lane == 1)  { a[0] = wsum[4]; a[1] = wsum[5]; }
        else if (lane == 17) { a[0] = wsum[6]; a[1] = wsum[7]; }
        v2f b; b[0] = 1.0f; b[1] = 1.0f;
        v8f c = {};
        c = __builtin_amdgcn_wmma_f32_16x16x4_f32(false, a, false, b,
                                                  (short)0, c, false, false);
        blocksum = c[0] + c[1];
#else
        if (lane == 0) {
            float s = 0.0f;
            #pragma unroll
            for (int i = 0; i < 8; ++i) s += wsum[i];
            blocksum = s;
        }
#endif
    }
    return blocksum;  // valid on thread 0
}

// ---------------- per-cell loss (pointers may be LDS or global) ----------------
__device__ __forceinline__ float cell_loss(const float* __restrict__ pcell,
                                           const float* __restrict__ tcell) {
    const v2f* pp = reinterpret_cast<const v2f*>(pcell);  // 120 B, 8 B aligned
    const v2f* tp = reinterpret_cast<const v2f*>(tcell);
    float p[30], t[30];
    #pragma unroll
    for (int i = 0; i < 15; ++i) { v2f v = pp[i]; p[2*i] = v[0]; p[2*i+1] = v[1]; }
    #pragma unroll
    for (int i = 0; i < 15; ++i) { v2f v = tp[i]; t[2*i] = v[0]; t[2*i+1] = v[1]; }

    const float t4  = t[4];
    const float coo = (t4 == 1.0f) ? 1.0f : 0.0f;
    const float noo = (t4 == 0.0f) ? 1.0f : 0.0f;

    const float dn0 = p[4] - t[4];
    const float dn1 = p[9] - t[9];
    const float nooobj = dn0 * dn0 + dn1 * dn1;

    float cls = 0.0f;
    #pragma unroll
    for (int i = 10; i < 30; ++i) { const float d = p[i] - t[i]; cls += d * d; }

    const float txc = t[0] / 7.0f, tyc = t[1] / 7.0f;
    const float tw  = t[2],        th  = t[3];
    const float tx0 = txc - 0.5f * tw, ty0 = tyc - 0.5f * th;
    const float tx1 = txc + 0.5f * tw, ty1 = tyc + 0.5f * th;
    const float a2  = (tx1 - tx0) * (ty1 - ty0);

    float iou[2], pc[2], bx[2], by[2], bw[2], bh[2];
    #pragma unroll
    for (int b = 0; b < 2; ++b) {
        const float px = p[5*b + 0], py = p[5*b + 1];
        const float pw = p[5*b + 2], ph = p[5*b + 3];
        bx[b] = px; by[b] = py; bw[b] = pw; bh[b] = ph; pc[b] = p[5*b + 4];
        const float pxc = px / 7.0f, pyc = py / 7.0f;
        const float px0 = pxc - 0.5f * pw, py0 = pyc - 0.5f * ph;
        const float px1 = pxc + 0.5f * pw, py1 = pyc + 0.5f * ph;
        const float ltx = fmaxf(px0, tx0), lty = fmaxf(py0, ty0);
        const float rbx = fminf(px1, tx1), rby = fminf(py1, ty1);
        const float wx  = fmaxf(rbx - ltx, 0.0f);
        const float wy  = fmaxf(rby - lty, 0.0f);
        const float inter = wx * wy;
        const float a1  = (px1 - px0) * (py1 - py0);
        iou[b] = inter / (a1 + a2);
    }

    const int   idx     = (iou[1] > iou[0]) ? 1 : 0;   // jnp.argmax tie -> 0
    const float max_iou = fmaxf(iou[0], iou[1]);

    const float rpc = pc[idx];
    const float npc = pc[1 - idx];
    float contain = rpc - max_iou; contain *= contain;

    const float dx = bx[idx] - t[0];
    const float dy = by[idx] - t[1];
    const float dw = sqrtf(bw[idx]) - sqrtf(t[2]);
    const float dh = sqrtf(bh[idx]) - sqrtf(t[3]);
    const float loc = dx * dx + dy * dy + dw * dw + dh * dh;
    const float not_contain = npc * npc;

    return coo * (5.0f * loc + 2.0f * contain + not_contain + cls)
         + noo * 0.5f * nooobj;
}

// ---------------- stage 1 ----------------
__global__ void __launch_bounds__(S1_BLOCK)
yolo_loss_partial(const float* __restrict__ pred,
                  const float* __restrict__ targ,
                  float* __restrict__ partials) {
    float acc = 0.0f;
#if USE_ASYNC_LDS
    // Double-buffered chunks: 128 cells x 240 B = 30720 B per buffer.
    __shared__ __align__(16) float lbuf[2][CHUNK_F4 * 4];
    const int tid = threadIdx.x;
    const long long cell0 = (long long)blockIdx.x * (CHUNK_CELLS * CHUNKS_PER_BLOCK);

    auto issue_chunk = [&](int chunk, int buf) {
        const float* predc = pred + (size_t)(cell0 + (long long)chunk * CHUNK_CELLS) * 30;
        const float* targc = targ + (size_t)(cell0 + (long long)chunk * CHUNK_CELLS) * 30;
        #pragma unroll
        for (int k = 0; k < ASYNC_ITERS; ++k) {
            const int idx = k * S1_BLOCK + tid;  // float4 slot 0..1919
            const float* g = (idx < CHUNK_F4_PRED)
                               ? (predc + (size_t)idx * 4)
                               : (targc + (size_t)(idx - CHUNK_F4_PRED) * 4);
            as1_v4i* gp = (as1_v4i*)(unsigned long long)(uintptr_t)g;
            as3_v4i* lp = (as3_v4i*)(unsigned)(uintptr_t)(&lbuf[buf][idx * 4]);
            __builtin_amdgcn_global_load_async_to_lds_b128(gp, lp, 0, 0);
        }
    };

    issue_chunk(0, 0);
    #pragma unroll
    for (int c = 0; c < CHUNKS_PER_BLOCK; ++c) {
        if (c + 1 < CHUNKS_PER_BLOCK) {
            issue_chunk(c + 1, (c + 1) & 1);
            // newest 15 in-flight = prefetch; async loads complete in order,
            // so <=15 outstanding guarantees chunk c has fully landed.
            __builtin_amdgcn_s_wait_asynccnt(ASYNC_ITERS);
        } else {
            __builtin_amdgcn_s_wait_asynccnt(0);
        }
        __syncthreads();  // cross-wave visibility of chunk c
        const float* pcell = &lbuf[c & 1][tid * 30];
        const float* tcell = &lbuf[c & 1][CHUNK_F4_PRED * 4 + tid * 30];
        acc += cell_loss(pcell, tcell);
        __syncthreads();  // all reads done before this buffer is overwritten
    }
#else
    // Fallback: direct strided global loads.
    const int tid = blockIdx.x * S1_BLOCK + threadIdx.x;
    const int T   = S1_GRID * S1_BLOCK;  // 100352
    #pragma unroll
    for (int c = 0; c < CHUNKS_PER_BLOCK; ++c) {
        const size_t cell = (size_t)tid + (size_t)c * (size_t)T;
        acc += cell_loss(pred + cell * 30, targ + cell * 30);
    }
#endif
    const float bs = block_reduce<S1_BLOCK / 32>(acc);
    if (threadIdx.x == 0) partials[blockIdx.x] = bs;
}

// ---------------- stage 2 ----------------
__global__ void __launch_bounds__(S2_BLOCK)
yolo_loss_final(const float* __restrict__ partials,
                float* __restrict__ out, int n) {
    float acc = 0.0f;
    for (int i = threadIdx.x; i < n; i += S2_BLOCK) acc += partials[i];  // fixed order
    const float bs = block_reduce<S2_BLOCK / 32>(acc);
    if (threadIdx.x == 0) out[0] = bs / 16384.0f;
}

extern "C" void kernel_launch(void* const* d_in, const int* in_sizes, int n_in,
                              void* d_out, int out_size, void* d_ws, size_t ws_size,
                              hipStream_t stream) {
    const float* pred = (const float*)d_in[0];   // (16384,7,7,30) f32
    const float* targ = (const float*)d_in[1];   // (16384,7,7,30) f32
    float* partials = (float*)d_ws;              // 784 floats of scratch
    yolo_loss_partial<<<S1_GRID, S1_BLOCK, 0, stream>>>(pred, targ, partials);
    yolo_loss_final<<<1, S2_BLOCK, 0, stream>>>(partials, (float*)d_out, S1_GRID);
}